// LSTM_6519760355926
// MI455X (gfx1250) — compile-verified
//
#include <hip/hip_runtime.h>

// CDNA5 / gfx1250: wave32. One wave runs the whole 3-layer LSTM recurrence
// in registers; final 6-wide FC dot is done with chained V_WMMA_F32_16X16X4_F32.

typedef float v2f __attribute__((ext_vector_type(2)));
typedef float v8f __attribute__((ext_vector_type(8)));

#define HN 6
#define NG 4

__device__ __forceinline__ float sigmoidf_fast(float x) {
    return 1.0f / (1.0f + __expf(-x));
}
// tanh(x) = 1 - 2/(e^{2x}+1): saturates cleanly to +-1 for large |x| (no inf/inf).
__device__ __forceinline__ float tanhf_fast(float x) {
    float e = __expf(2.0f * x);
    return 1.0f - 2.0f / (e + 1.0f);
}

__global__ __launch_bounds__(32, 1)
void lstm3_seq_wmma_kernel(const float* __restrict__ x,
                           const float* __restrict__ wih0, const float* __restrict__ whh0,
                           const float* __restrict__ bih0, const float* __restrict__ bhh0,
                           const float* __restrict__ wih1, const float* __restrict__ whh1,
                           const float* __restrict__ bih1, const float* __restrict__ bhh1,
                           const float* __restrict__ wih2, const float* __restrict__ whh2,
                           const float* __restrict__ bih2, const float* __restrict__ bhh2,
                           const float* __restrict__ fcw, const float* __restrict__ fcb,
                           float* __restrict__ out, int T)
{
    const int lane = (int)threadIdx.x;        // 0..31, single wave
    const int j = (lane < HN) ? lane : 0;     // unit index; lanes >=6 shadow unit 0

    // Lane j owns rows {k*H + j} for gates k = i,f,g,o of every layer.
    float Wi0[NG];                 // layer0 input weight (input size 1)
    float Wh0[NG][HN], B0[NG];
    float Wi1[NG][HN], Wh1[NG][HN], B1[NG];
    float Wi2[NG][HN], Wh2[NG][HN], B2[NG];

#pragma unroll
    for (int k = 0; k < NG; ++k) {
        const int row = k * HN + j;
        Wi0[k] = wih0[row];                        // (4H,1)
        B0[k]  = bih0[row] + bhh0[row];
        B1[k]  = bih1[row] + bhh1[row];
        B2[k]  = bih2[row] + bhh2[row];
#pragma unroll
        for (int c = 0; c < HN; ++c) {
            Wh0[k][c] = whh0[row * HN + c];
            Wi1[k][c] = wih1[row * HN + c];
            Wh1[k][c] = whh1[row * HN + c];
            Wi2[k][c] = wih2[row * HN + c];
            Wh2[k][c] = whh2[row * HN + c];
        }
    }

    // Wave-uniform hidden state (readlane-broadcast each step); cell state per lane.
    float h0[HN], h1[HN], h2[HN];
#pragma unroll
    for (int c = 0; c < HN; ++c) { h0[c] = 0.0f; h1[c] = 0.0f; h2[c] = 0.0f; }
    float c0 = 0.0f, c1 = 0.0f, c2 = 0.0f;

    float xcur = x[0];
    for (int t = 0; t < T; ++t) {
        const int tn = (t + 1 < T) ? (t + 1) : (T - 1);
        const float xnext = x[tn];               // prefetch; hidden under 3-layer compute

        float a[NG];

        // ---------------- layer 0 ----------------
#pragma unroll
        for (int k = 0; k < NG; ++k) {
            float acc = B0[k] + Wi0[k] * xcur;
#pragma unroll
            for (int c = 0; c < HN; ++c) acc += Wh0[k][c] * h0[c];
            a[k] = acc;
        }
        {
            const float ig = sigmoidf_fast(a[0]);
            const float fg = sigmoidf_fast(a[1]);
            const float gg = tanhf_fast(a[2]);
            const float og = sigmoidf_fast(a[3]);
            c0 = fg * c0 + ig * gg;
            const float hme = og * tanhf_fast(c0);
#pragma unroll
            for (int c = 0; c < HN; ++c) h0[c] = __builtin_amdgcn_readlane(hme, c);
        }

        // ---------------- layer 1 ----------------
#pragma unroll
        for (int k = 0; k < NG; ++k) {
            float acc = B1[k];
#pragma unroll
            for (int c = 0; c < HN; ++c) acc += Wi1[k][c] * h0[c] + Wh1[k][c] * h1[c];
            a[k] = acc;
        }
        {
            const float ig = sigmoidf_fast(a[0]);
            const float fg = sigmoidf_fast(a[1]);
            const float gg = tanhf_fast(a[2]);
            const float og = sigmoidf_fast(a[3]);
            c1 = fg * c1 + ig * gg;
            const float hme = og * tanhf_fast(c1);
#pragma unroll
            for (int c = 0; c < HN; ++c) h1[c] = __builtin_amdgcn_readlane(hme, c);
        }

        // ---------------- layer 2 ----------------
#pragma unroll
        for (int k = 0; k < NG; ++k) {
            float acc = B2[k];
#pragma unroll
            for (int c = 0; c < HN; ++c) acc += Wi2[k][c] * h1[c] + Wh2[k][c] * h2[c];
            a[k] = acc;
        }
        {
            const float ig = sigmoidf_fast(a[0]);
            const float fg = sigmoidf_fast(a[1]);
            const float gg = tanhf_fast(a[2]);
            const float og = sigmoidf_fast(a[3]);
            c2 = fg * c2 + ig * gg;
            const float hme = og * tanhf_fast(c2);
#pragma unroll
            for (int c = 0; c < HN; ++c) h2[c] = __builtin_amdgcn_readlane(hme, c);
        }

        xcur = xnext;
    }

    // ---------------- final FC via V_WMMA_F32_16X16X4_F32 ----------------
    // out = sum_j fc_w[j]*h2[j] + fc_b, computed as D = A(16x4) x B(4x16) + C,
    // reading D[0][0]. A row 0 holds h2, B column 0 holds fc_w, split K=6 into 4+2.
    // A layout: lanes 0-15 -> K=0,1 in the two VGPRs; lanes 16-31 -> K=2,3.
    // B layout: lanes 0-15 (N=0..15) -> rows K=0,1; lanes 16-31 -> rows K=2,3.
    const float f0 = fcw[0], f1 = fcw[1], f2 = fcw[2];
    const float f3 = fcw[3], f4 = fcw[4], f5 = fcw[5];
    const bool l0 = (lane == 0), l16 = (lane == 16);

    v2f A0, Bm0, A1, Bm1;
    A0.x  = l0 ? h2[0] : (l16 ? h2[2] : 0.0f);
    A0.y  = l0 ? h2[1] : (l16 ? h2[3] : 0.0f);
    Bm0.x = l0 ? f0    : (l16 ? f2    : 0.0f);
    Bm0.y = l0 ? f1    : (l16 ? f3    : 0.0f);
    A1.x  = l0 ? h2[4] : 0.0f;
    A1.y  = l0 ? h2[5] : 0.0f;
    Bm1.x = l0 ? f4    : 0.0f;
    Bm1.y = l0 ? f5    : 0.0f;

    v8f acc8 = {};
    acc8 = __builtin_amdgcn_wmma_f32_16x16x4_f32(false, A0, false, Bm0,
                                                 (short)0, acc8, false, false);
    acc8 = __builtin_amdgcn_wmma_f32_16x16x4_f32(false, A1, false, Bm1,
                                                 (short)0, acc8, false, false);

    if (lane == 0) {
        out[0] = acc8[0] + fcb[0];   // D[0][0] lives in lane 0, VGPR 0 of C/D tile
    }
}

extern "C" void kernel_launch(void* const* d_in, const int* in_sizes, int n_in,
                              void* d_out, int out_size, void* d_ws, size_t ws_size,
                              hipStream_t stream) {
    const float* x    = (const float*)d_in[0];
    const float* wih0 = (const float*)d_in[1];
    const float* whh0 = (const float*)d_in[2];
    const float* bih0 = (const float*)d_in[3];
    const float* bhh0 = (const float*)d_in[4];
    const float* wih1 = (const float*)d_in[5];
    const float* whh1 = (const float*)d_in[6];
    const float* bih1 = (const float*)d_in[7];
    const float* bhh1 = (const float*)d_in[8];
    const float* wih2 = (const float*)d_in[9];
    const float* whh2 = (const float*)d_in[10];
    const float* bih2 = (const float*)d_in[11];
    const float* bhh2 = (const float*)d_in[12];
    const float* fcw  = (const float*)d_in[13];
    const float* fcb  = (const float*)d_in[14];
    float* out = (float*)d_out;

    const int T = in_sizes[0];   // x is (1, T, 1) -> 65536 elements

    // Inherently serial recurrence: one wave32 does the whole thing in registers.
    lstm3_seq_wmma_kernel<<<1, 32, 0, stream>>>(
        x, wih0, whh0, bih0, bhh0,
        wih1, whh1, bih1, bhh1,
        wih2, whh2, bih2, bhh2,
        fcw, fcb, out, T);
}